// AGREE_1769526526109
// MI455X (gfx1250) — compile-verified
//
#include <hip/hip_runtime.h>

typedef __attribute__((ext_vector_type(2))) float v2f;
typedef __attribute__((ext_vector_type(8))) float v8f;

#define DD   256   // embedding dim
#define MM   50    // members per group
#define MPAD 64    // padded to 4 WMMA row-tiles

// LDS layout (floats)
#define OFF_X      0                         // MPAD*DD      = 16384
#define OFF_W1S    (OFF_X + MPAD*DD)         // 4096 (permuted att_w1[0:256,:] B-fragments)
#define OFF_HT     (OFF_W1S + 4096)          // MPAD*16      = 1024
#define OFF_ITEM   (OFF_HT + MPAD*16)        // 256
#define OFF_GROUP  (OFF_ITEM + DD)           // 256
#define OFF_PART   (OFF_GROUP + DD)          // 256 scratch (reused)
#define OFF_NEW    (OFF_PART + 256)          // 768
#define OFF_IW1    (OFF_NEW + 3*DD)          // 16
#define OFF_B1     (OFF_IW1 + 16)            // 16
#define OFF_W2     (OFF_B1 + 16)             // 16
#define OFF_MASK   (OFF_W2 + 16)             // 64
#define OFF_E      (OFF_MASK + MPAD)         // 64
#define OFF_RED    (OFF_E + MPAD)            // 4
#define OFF_IDS    (OFF_RED + 4)             // 64 (ints)
#define SMEM_FLOATS (OFF_IDS + 64)

__global__ __launch_bounds__(256) void agree_fused(
    const int*   __restrict__ user_inputs,
    const int*   __restrict__ item_inputs,
    const int*   __restrict__ menb_ids,
    const float* __restrict__ mask,
    const float* __restrict__ user_table,
    const float* __restrict__ item_table,
    const float* __restrict__ group_table,
    const float* __restrict__ att_w1,
    const float* __restrict__ att_b1,
    const float* __restrict__ att_w2,
    const float* __restrict__ att_b2,
    const float* __restrict__ pred_w1,
    const float* __restrict__ pred_b1,
    const float* __restrict__ pred_w2,
    const float* __restrict__ pred_b2,
    float* __restrict__ out)
{
  extern __shared__ float smem[];
  float* X      = smem + OFF_X;
  float* W1s    = smem + OFF_W1S;
  float* Ht     = smem + OFF_HT;
  float* item_s = smem + OFF_ITEM;
  float* group_s= smem + OFF_GROUP;
  float* part   = smem + OFF_PART;
  float* new_s  = smem + OFF_NEW;
  float* iw1    = smem + OFF_IW1;
  float* b1s    = smem + OFF_B1;
  float* w2s    = smem + OFF_W2;
  float* mask_s = smem + OFF_MASK;
  float* e_s    = smem + OFF_E;
  float* red    = smem + OFF_RED;
  int*   ids_s  = (int*)(smem + OFF_IDS);

  const int t = threadIdx.x;
  const int b = blockIdx.x;

  // ---------------- stage: ids, mask, item, group, biases, permuted W1 ----
  if (t < MPAD) {
    if (t < MM) { ids_s[t] = menb_ids[b*MM + t]; mask_s[t] = mask[b*MM + t]; }
    else        { ids_s[t] = 0;                  mask_s[t] = 0.f; }
  }
  if (t < 16) { b1s[t] = att_b1[t]; w2s[t] = att_w2[t]; }
  {
    const int item_id = item_inputs[b];
    const int grp     = user_inputs[b];
    item_s[t]  = item_table [(size_t)item_id * DD + t];
    group_s[t] = group_table[(size_t)grp     * DD + t];
  }
  // att_w1[0:256,:] into WMMA-B fragment order:
  // pair p = g*32 + lane  ->  { W1[4g+2*(lane>>4)][lane&15], W1[4g+2*(lane>>4)+1][lane&15] }
  #pragma unroll
  for (int i = 0; i < 8; ++i) {
    int p = t + (i << 8);
    int l = p & 31;
    int n = l & 15;
    int k = 4*(p >> 5) + 2*(l >> 4);
    W1s[2*p]     = att_w1[k*16 + n];
    W1s[2*p + 1] = att_w1[(k+1)*16 + n];
  }
  __syncthreads();

  // ---------------- gather masked member embeddings into X (f32, b128) ----
  for (int p = 0; p < 13; ++p) {
    int m  = p*4 + (t >> 6);
    int c4 = (t & 63) * 4;
    if (m < MM) {
      const float4 v = *(const float4*)(user_table + (size_t)ids_s[m] * DD + c4);
      const float mk = mask_s[m];
      float* dst = X + m*DD + c4;
      dst[0] = v.x*mk; dst[1] = v.y*mk; dst[2] = v.z*mk; dst[3] = v.w*mk;
    } else if (m < 52) {
      float* dst = X + m*DD + c4;
      dst[0] = 0.f; dst[1] = 0.f; dst[2] = 0.f; dst[3] = 0.f;
    }
  }
  for (int m = 52; m < MPAD; ++m) X[m*DD + t] = 0.f;

  // ---------------- iw1[n] = item . att_w1[256:512, n] (rank-1 fold) ------
  {
    const int j = t & 15, c = t >> 4;
    float s = 0.f;
    #pragma unroll
    for (int i = 0; i < 16; ++i) {
      const int ii = c*16 + i;
      s += item_s[ii] * att_w1[(DD + ii)*16 + j];
    }
    part[t] = s;
  }
  __syncthreads();
  if (t < 16) {
    float s = 0.f;
    for (int c = 0; c < 16; ++c) s += part[c*16 + t];
    iw1[t] = s;
  }
  __syncthreads();

  // ---------------- attention GEMM: [64x256] x [256x16] via WMMA f32 -----
  const int wave = t >> 5;
  const int lane = t & 31;
  if (wave < 4) {
    const int rbase = wave << 4;
    const int kh    = lane >> 4;
    const int n     = lane & 15;
    const float* Arow = X + (rbase + n)*DD + 2*kh;  // row = rbase + (lane&15)
    const float* Bp   = W1s + 2*lane;
    v8f acc = {};
    #pragma unroll 8
    for (int g = 0; g < 64; ++g) {
      v2f a  = *(const v2f*)(Arow + 4*g);
      v2f bb = *(const v2f*)(Bp + 64*g);
      acc = __builtin_amdgcn_wmma_f32_16x16x4_f32(
          false, a, false, bb, (short)0, acc, false, false);
    }
    const float bias = b1s[n];
    const float iwn  = iw1[n];
    #pragma unroll
    for (int r = 0; r < 8; ++r) {
      const int m = rbase + r + 8*kh;      // C/D layout: M = r (+8 for hi half)
      const float h = acc[r] + bias + mask_s[m]*iwn;
      Ht[m*16 + n] = fmaxf(h, 0.f);
    }
  }
  __syncthreads();

  // ---------------- logits, clip, masked softmax --------------------------
  if (t < MPAD) {
    float e = 0.f;
    if (t < MM) {
      float a = att_b2[0];
      #pragma unroll
      for (int n2 = 0; n2 < 16; ++n2) a += Ht[t*16 + n2] * w2s[n2];
      a = fminf(50.f, fmaxf(-50.f, a));
      e = __expf(a) * mask_s[t];
    }
    e_s[t] = e;
  }
  __syncthreads();
  if (t == 0) {
    float s = 0.f;
    for (int m = 0; m < MM; ++m) s += e_s[m];
    red[0] = 1.f / s;
  }
  __syncthreads();

  // ---------------- weighted pooling + build new = [g*item, g, item] ------
  {
    float acc = 0.f;
    for (int m = 0; m < MM; ++m) acc += e_s[m] * X[m*DD + t];
    acc *= red[0];                       // / sum(e)
    const float g  = 0.5f*acc + group_s[t];   // LMD = 0.5
    const float it = item_s[t];
    new_s[t]        = g * it;
    new_s[DD + t]   = g;
    new_s[2*DD + t] = it;
  }
  __syncthreads();

  // ---------------- predict MLP: relu(new@W1p+b1)@w2p + b2 ----------------
  if (t < 64) {
    const int j = t & 7, ch = t >> 3;
    float s = 0.f;
    for (int i = ch*96; i < ch*96 + 96; ++i) s += new_s[i] * pred_w1[i*8 + j];
    part[t] = s;
  }
  __syncthreads();
  if (t < 8) {
    float s = pred_b1[t];
    for (int c2 = 0; c2 < 8; ++c2) s += part[c2*8 + t];
    part[64 + t] = fmaxf(s, 0.f) * pred_w2[t];
  }
  __syncthreads();
  if (t == 0) {
    float o = pred_b2[0];
    for (int j2 = 0; j2 < 8; ++j2) o += part[64 + j2];
    out[b] = o;
  }
}

extern "C" void kernel_launch(void* const* d_in, const int* in_sizes, int n_in,
                              void* d_out, int out_size, void* d_ws, size_t ws_size,
                              hipStream_t stream) {
  (void)n_in; (void)out_size; (void)d_ws; (void)ws_size;
  const int B = in_sizes[0];
  constexpr size_t SMEM = (size_t)SMEM_FLOATS * sizeof(float);
  static_assert(SMEM < 320u*1024u, "fits WGP LDS");
  // > 64KB dynamic LDS: raise the limit (no-op if already allowed)
  (void)hipFuncSetAttribute((const void*)agree_fused,
                            hipFuncAttributeMaxDynamicSharedMemorySize, (int)SMEM);
  agree_fused<<<B, 256, SMEM, stream>>>(
      (const int*)d_in[0],  (const int*)d_in[1],  (const int*)d_in[2],
      (const float*)d_in[3], (const float*)d_in[4], (const float*)d_in[5],
      (const float*)d_in[6], (const float*)d_in[7], (const float*)d_in[8],
      (const float*)d_in[9], (const float*)d_in[10], (const float*)d_in[11],
      (const float*)d_in[12], (const float*)d_in[13], (const float*)d_in[14],
      (float*)d_out);
}